// MAE_MaskingRaw_12352325943963
// MI455X (gfx1250) — compile-verified
//
#include <hip/hip_runtime.h>
#include <cstdint>

#define Bb 32
#define Tt 2048
#define Kk 512
#define MM 1024
#define MU 1024
#define LN_EPS 1e-5f

// Native clang vector type: legal for __builtin_nontemporal_store (HIP's
// float4 is a class and is rejected by the builtin).
typedef float v4f __attribute__((ext_vector_type(4)));

// ---------------------------------------------------------------------------
// Kernel A: streaming row gather via CDNA5 async global<->LDS DMA.
// One block (128 lanes / 4 waves) per (b, j) output row pair:
//   out_emb[b,j,:] = token_embedding[b, unmask_index[b,j], :]  (or 0)
//   out_pos[b,j,:] = pe[unmask_index[b,j], :]                   (or 0)
// Each lane moves 16 bytes; each wave issues 2 async b128 loads into LDS,
// waits ASYNCcnt==0, then 2 async b128 stores from LDS. All per-wave, so no
// cross-wave LDS synchronization is needed. s_endpgm drains remaining stores.
//
// Cache policy (MI455X, 192 MB L2, 23.3 TB/s HBM):
//   - token_embedding rows are read exactly once  -> TH_LOAD_NT
//   - pe (4 MiB) is re-gathered ~32x across batch -> RT (keep L2-resident)
//   - outputs are write-once streams              -> TH_STORE_NT / nt stores
// ---------------------------------------------------------------------------
__global__ __launch_bounds__(128) void mae_gather_async_kernel(
    const float* __restrict__ tok,
    const float* __restrict__ pe,
    const int*   __restrict__ unmask_index,
    const int*   __restrict__ num_unmasked,
    float* __restrict__ out_emb,
    float* __restrict__ out_pos)
{
    __shared__ __align__(16) float lds[2 * Kk];   // 4 KB tile: [tok row | pe row]

    const int row  = blockIdx.x;          // b*MU + j
    const int b    = row >> 10;
    const int j    = row & (MU - 1);
    const int lane = threadIdx.x;         // 0..127, owns 4 floats (16 B)

    float* o1 = out_emb + (size_t)row * Kk + lane * 4;
    float* o3 = out_pos + (size_t)row * Kk + lane * 4;

    const bool valid = j < num_unmasked[b];   // uniform across the block

    if (valid) {
        // Block-uniform gather index -> scalar register.
        const int idx = __builtin_amdgcn_readfirstlane(unmask_index[row]);

        const float* gt = tok + ((size_t)b * Tt + (size_t)idx) * Kk + lane * 4;
        const float* gp = pe  + (size_t)idx * Kk + lane * 4;

        // LDS byte address: low 32 bits of the generic pointer hold the LDS
        // offset (aperture tag lives in the high word).
        const unsigned ldsT = (unsigned)(uintptr_t)(&lds[0]) + (unsigned)(lane * 16);
        const unsigned ldsP = ldsT + (unsigned)(Kk * 4);

        // token row: read-once -> non-temporal
        asm volatile("global_load_async_to_lds_b128 %0, %1, off th:TH_LOAD_NT"
                     :: "v"(ldsT), "v"((uint64_t)(uintptr_t)gt) : "memory");
        // pe row: heavily reused across the batch -> default RT (L2-resident)
        asm volatile("global_load_async_to_lds_b128 %0, %1, off"
                     :: "v"(ldsP), "v"((uint64_t)(uintptr_t)gp) : "memory");

        asm volatile("s_wait_asynccnt 0x0" ::: "memory");

        // outputs: write-once streams -> non-temporal stores
        asm volatile("global_store_async_from_lds_b128 %0, %1, off th:TH_STORE_NT"
                     :: "v"((uint64_t)(uintptr_t)o1), "v"(ldsT) : "memory");
        asm volatile("global_store_async_from_lds_b128 %0, %1, off th:TH_STORE_NT"
                     :: "v"((uint64_t)(uintptr_t)o3), "v"(ldsP) : "memory");
        // ASYNCcnt drained by implicit s_wait_idle at s_endpgm.
    } else {
        const v4f z = {0.f, 0.f, 0.f, 0.f};
        __builtin_nontemporal_store(z, (v4f*)o1);
        __builtin_nontemporal_store(z, (v4f*)o3);
    }
}

// ---------------------------------------------------------------------------
// Kernel B: mask_embedding[b,j,:] = LN(mask_vec + pe[mask_index[b,j]]) or 0.
// 128 lanes per row; wave32 xor-reduction via ds_swizzle, 4-wave LDS combine.
// ---------------------------------------------------------------------------
#define SWZ_ADD(v, imm) \
    (v) += __int_as_float(__builtin_amdgcn_ds_swizzle(__float_as_int(v), (imm)))

__global__ __launch_bounds__(128) void mae_mask_ln_kernel(
    const float* __restrict__ pe,
    const float* __restrict__ mw,       // mask_weight[:,0], K floats
    const float* __restrict__ gamma,
    const float* __restrict__ beta,
    const int*   __restrict__ mask_index,
    const int*   __restrict__ num_masked,
    float* __restrict__ out)
{
    __shared__ float red_s[4];
    __shared__ float red_q[4];

    const int row  = blockIdx.x;          // b*MM + j
    const int b    = row >> 10;
    const int j    = row & (MM - 1);
    const int lane = threadIdx.x;         // 0..127
    const int wave = lane >> 5;

    float* o = out + (size_t)row * Kk + lane * 4;

    const bool valid = j < num_masked[b];   // uniform across the block
    if (!valid) {                           // uniform exit: no barrier split
        const v4f z = {0.f, 0.f, 0.f, 0.f};
        __builtin_nontemporal_store(z, (v4f*)o);
        return;
    }

    const int idx = __builtin_amdgcn_readfirstlane(mask_index[row]);

    const v4f pv = *(const v4f*)(pe + (size_t)idx * Kk + lane * 4);
    const v4f wv = *(const v4f*)(mw + lane * 4);

    const float x0 = pv.x + wv.x;
    const float x1 = pv.y + wv.y;
    const float x2 = pv.z + wv.z;
    const float x3 = pv.w + wv.w;

    float s = x0 + x1 + x2 + x3;
    float q = x0 * x0 + x1 * x1 + x2 * x2 + x3 * x3;

    // wave32 butterfly reduction (lane ^ 16, 8, 4, 2, 1)
    SWZ_ADD(s, 0x401F); SWZ_ADD(q, 0x401F);
    SWZ_ADD(s, 0x201F); SWZ_ADD(q, 0x201F);
    SWZ_ADD(s, 0x101F); SWZ_ADD(q, 0x101F);
    SWZ_ADD(s, 0x081F); SWZ_ADD(q, 0x081F);
    SWZ_ADD(s, 0x041F); SWZ_ADD(q, 0x041F);

    if ((lane & 31) == 0) { red_s[wave] = s; red_q[wave] = q; }
    __syncthreads();

    s = red_s[0] + red_s[1] + red_s[2] + red_s[3];
    q = red_q[0] + red_q[1] + red_q[2] + red_q[3];

    const float mean = s * (1.0f / Kk);
    const float var  = q * (1.0f / Kk) - mean * mean;
    const float r    = rsqrtf(var + LN_EPS);

    const v4f g  = *(const v4f*)(gamma + lane * 4);
    const v4f be = *(const v4f*)(beta  + lane * 4);

    v4f y;
    y.x = (x0 - mean) * r * g.x + be.x;
    y.y = (x1 - mean) * r * g.y + be.y;
    y.z = (x2 - mean) * r * g.z + be.z;
    y.w = (x3 - mean) * r * g.w + be.w;
    __builtin_nontemporal_store(y, (v4f*)o);
}

extern "C" void kernel_launch(void* const* d_in, const int* in_sizes, int n_in,
                              void* d_out, int out_size, void* d_ws, size_t ws_size,
                              hipStream_t stream) {
    const float* tok   = (const float*)d_in[0];   // (B,T,K)
    const float* pe    = (const float*)d_in[1];   // (T,K)
    const float* mw    = (const float*)d_in[2];   // (K,1)
    const float* gam   = (const float*)d_in[3];   // (K,)
    const float* bet   = (const float*)d_in[4];   // (K,)
    const int*   midx  = (const int*)d_in[5];     // (B,MM)
    const int*   uidx  = (const int*)d_in[6];     // (B,MU)
    const int*   nm    = (const int*)d_in[7];     // (B,)
    const int*   nu    = (const int*)d_in[8];     // (B,)

    float* out  = (float*)d_out;
    float* out1 = out;                                    // unmasked_embeddings
    float* out2 = out1 + (size_t)Bb * MU * Kk;            // mask_embedding
    float* out3 = out2 + (size_t)Bb * MM * Kk;            // unmasked_positions

    mae_gather_async_kernel<<<Bb * MU, 128, 0, stream>>>(tok, pe, uidx, nu, out1, out3);
    mae_mask_ln_kernel<<<Bb * MM, 128, 0, stream>>>(pe, mw, gam, bet, midx, nm, out2);
}